// pre_embedding_47382079209473
// MI455X (gfx1250) — compile-verified
//
#include <hip/hip_runtime.h>
#include <hip/hip_bf16.h>

typedef __attribute__((ext_vector_type(2))) float v2f;
typedef __attribute__((ext_vector_type(8))) float v8f;

#define GCN_N    65536
#define GCN_E    524288
#define GCN_D    128
#define GCN_G    512
#define GCN_NC   10

// ---------------------------------------------------------------------------
// generic zero-fill (scalar float grid-stride)
__global__ void gcn_zero_kernel(float* __restrict__ p, int n_floats) {
    int i = blockIdx.x * blockDim.x + threadIdx.x;
    int stride = gridDim.x * blockDim.x;
    for (int j = i; j < n_floats; j += stride) p[j] = 0.0f;
}

// ---------------------------------------------------------------------------
// degree accumulation: deg_out[src]++ , deg_in[dst]++
__global__ void gcn_degree_kernel(const int* __restrict__ src,
                                  const int* __restrict__ dst,
                                  float* __restrict__ deg_out,
                                  float* __restrict__ deg_in, int e) {
    int i = blockIdx.x * blockDim.x + threadIdx.x;
    if (i >= e) return;
    atomicAdd(&deg_out[src[i]], 1.0f);
    atomicAdd(&deg_in[dst[i]], 1.0f);
}

// in-place deg -> rsqrt(max(deg,1))
__global__ void gcn_norm_kernel(float* __restrict__ a, float* __restrict__ b, int n) {
    int i = blockIdx.x * blockDim.x + threadIdx.x;
    if (i >= n) return;
    a[i] = rsqrtf(fmaxf(a[i], 1.0f));
    b[i] = rsqrtf(fmaxf(b[i], 1.0f));
}

// ---------------------------------------------------------------------------
// edge scatter: agg[dst] += x[src] * norm_src[src]
// one wave (32 lanes) per edge, each lane owns 4 contiguous features (float4)
__global__ void gcn_scatter_kernel(const float* __restrict__ x,
                                   const float* __restrict__ norm_src,
                                   const int* __restrict__ src,
                                   const int* __restrict__ dst,
                                   float* __restrict__ agg, int e) {
    int t = blockIdx.x * blockDim.x + threadIdx.x;
    int edge = t >> 5;
    if (edge >= e) return;
    int lane = t & 31;
    int s = src[edge];
    int d = dst[edge];
    float ns = norm_src[s];
    float4 v = ((const float4*)(x + (size_t)s * GCN_D))[lane];
    float* a = agg + (size_t)d * GCN_D + lane * 4;
    atomicAdd(a + 0, v.x * ns);
    atomicAdd(a + 1, v.y * ns);
    atomicAdd(a + 2, v.z * ns);
    atomicAdd(a + 3, v.w * ns);
}

// ---------------------------------------------------------------------------
// out = relu( (agg * norm_dst) @ W + b ) via V_WMMA_F32_16X16X4_F32
// grid: (N/16, D/16), block: 32 (one wave per 16x16 output tile)
// norm_dst row-scaling is algebraically hoisted to the epilogue.
__global__ __launch_bounds__(32)
void gcn_gemm_relu_kernel(const float* __restrict__ agg,
                          const float* __restrict__ norm_dst,
                          const float* __restrict__ W,   // [128,128] row-major
                          const float* __restrict__ bias,
                          float* __restrict__ out) {
    const int lane  = threadIdx.x;          // 0..31, all lanes active (EXEC==~0)
    const int row0  = blockIdx.x * 16;
    const int col0  = blockIdx.y * 16;
    const int m     = lane & 15;            // M (for A) / N (for B,C,D)
    const int khalf = (lane >> 4) * 2;      // lanes 0-15: K+0,K+1 ; 16-31: K+2,K+3

    const float* arow = agg + (size_t)(row0 + m) * GCN_D;

    v8f c = {0.f, 0.f, 0.f, 0.f, 0.f, 0.f, 0.f, 0.f};

#pragma unroll
    for (int k = 0; k < GCN_D; k += 4) {
        // A fragment: 16x4 f32, M=lane%16, K per ISA layout
        v2f a;
        const float2 av = *(const float2*)(arow + k + khalf);
        a.x = av.x;
        a.y = av.y;
        // B fragment: 4x16 f32, N=lane%16, K rows symmetric to A
        v2f bf;
        bf.x = W[(size_t)(k + khalf)     * GCN_D + col0 + m];
        bf.y = W[(size_t)(k + khalf + 1) * GCN_D + col0 + m];
        c = __builtin_amdgcn_wmma_f32_16x16x4_f32(
                /*neg_a=*/false, a, /*neg_b=*/false, bf,
                /*c_mod=*/(short)0, c, /*reuse_a=*/false, /*reuse_b=*/false);
    }

    // epilogue: D[j][lane] -> row = row0 + mbase + j, col = col0 + (lane&15)
    const int mbase = (lane >> 4) * 8;
    const float bv = bias[col0 + m];
#pragma unroll
    for (int j = 0; j < 8; ++j) {
        int r = row0 + mbase + j;
        float v = c[j] * norm_dst[r] + bv;
        v = v > 0.0f ? v : 0.0f;
        out[(size_t)r * GCN_D + col0 + m] = v;
    }
}

// ---------------------------------------------------------------------------
// per-graph sum pooling (+ count); one wave per node, lane owns float4
__global__ void gcn_pool_kernel(const float* __restrict__ h,
                                const int* __restrict__ graph_id,
                                float* __restrict__ pooled,
                                float* __restrict__ cnt, int n) {
    int t = blockIdx.x * blockDim.x + threadIdx.x;
    int node = t >> 5;
    if (node >= n) return;
    int lane = t & 31;
    int g = graph_id[node];
    float4 v = ((const float4*)(h + (size_t)node * GCN_D))[lane];
    float* p = pooled + (size_t)g * GCN_D + lane * 4;
    atomicAdd(p + 0, v.x);
    atomicAdd(p + 1, v.y);
    atomicAdd(p + 2, v.z);
    atomicAdd(p + 3, v.w);
    if (lane == 0) atomicAdd(&cnt[g], 1.0f);
}

// hidden[g][j] = relu( (pooled[g]/max(cnt,1)) . Wp1[:,j] + bp1[j] )
__global__ void gcn_head1_kernel(const float* __restrict__ pooled,
                                 const float* __restrict__ cnt,
                                 const float* __restrict__ Wp1,   // [128,64]
                                 const float* __restrict__ bp1,
                                 float* __restrict__ hidden) {
    int idx = blockIdx.x * blockDim.x + threadIdx.x;
    if (idx >= GCN_G * 64) return;
    int g = idx >> 6;
    int j = idx & 63;
    float inv = 1.0f / fmaxf(cnt[g], 1.0f);
    const float* pr = pooled + (size_t)g * GCN_D;
    float acc = bp1[j];
#pragma unroll 4
    for (int k = 0; k < GCN_D; ++k)
        acc += pr[k] * inv * Wp1[(size_t)k * 64 + j];
    hidden[idx] = acc > 0.0f ? acc : 0.0f;
}

// scores[g][c] = hidden[g] . Wp2[:,c] + bp2[c]
__global__ void gcn_head2_kernel(const float* __restrict__ hidden,
                                 const float* __restrict__ Wp2,   // [64,10]
                                 const float* __restrict__ bp2,
                                 float* __restrict__ scores) {
    int idx = blockIdx.x * blockDim.x + threadIdx.x;
    if (idx >= GCN_G * GCN_NC) return;
    int g = idx / GCN_NC;
    int cc = idx % GCN_NC;
    const float* hr = hidden + (size_t)g * 64;
    float acc = bp2[cc];
#pragma unroll 4
    for (int j = 0; j < 64; ++j)
        acc += hr[j] * Wp2[(size_t)j * GCN_NC + cc];
    scores[idx] = acc;
}

// ---------------------------------------------------------------------------
extern "C" void kernel_launch(void* const* d_in, const int* in_sizes, int n_in,
                              void* d_out, int out_size, void* d_ws, size_t ws_size,
                              hipStream_t stream) {
    (void)in_sizes; (void)n_in; (void)out_size; (void)ws_size;

    const float* hx       = (const float*)d_in[0];
    const int*   src      = (const int*)d_in[1];
    const int*   dst      = (const int*)d_in[2];
    const int*   graph_id = (const int*)d_in[3];
    // d_in[4] = num_graphs (compile-time 512 here)
    const float* Wl[6] = { (const float*)d_in[5],  (const float*)d_in[7],
                           (const float*)d_in[9],  (const float*)d_in[11],
                           (const float*)d_in[13], (const float*)d_in[15] };
    const float* bl[6] = { (const float*)d_in[6],  (const float*)d_in[8],
                           (const float*)d_in[10], (const float*)d_in[12],
                           (const float*)d_in[14], (const float*)d_in[16] };
    const float* Wp1 = (const float*)d_in[17];
    const float* bp1 = (const float*)d_in[18];
    const float* Wp2 = (const float*)d_in[19];
    const float* bp2 = (const float*)d_in[20];
    float* scores = (float*)d_out;

    // workspace layout
    char* ws = (char*)d_ws;
    float* normS  = (float*)(ws + 0x0000000);                 // N floats (deg_out -> norm_src)
    float* normD  = (float*)(ws + 0x0040000);                 // N floats (deg_in -> norm_dst)
    float* pooled = (float*)(ws + 0x0080000);                 // G*128
    float* cnt    = (float*)(ws + 0x00C0000);                 // G
    float* hidden = (float*)(ws + 0x00C1000);                 // G*64
    float* H0     = (float*)(ws + 0x0100000);                 // N*128
    float* H1     = (float*)(ws + 0x2100000);                 // N*128
    float* AGG    = (float*)(ws + 0x4100000);                 // N*128

    const int ZB = 256;

    // --- degrees -> norms ---------------------------------------------------
    gcn_zero_kernel<<<1024, ZB, 0, stream>>>(normS, GCN_N);
    gcn_zero_kernel<<<1024, ZB, 0, stream>>>(normD, GCN_N);
    gcn_degree_kernel<<<(GCN_E + ZB - 1) / ZB, ZB, 0, stream>>>(src, dst, normS, normD, GCN_E);
    gcn_norm_kernel<<<(GCN_N + ZB - 1) / ZB, ZB, 0, stream>>>(normS, normD, GCN_N);

    // --- 6 conv layers ------------------------------------------------------
    const float* layer_in = hx;
    float* ping[2] = { H0, H1 };
    dim3 gemm_grid(GCN_N / 16, GCN_D / 16);
    const int scatter_threads = GCN_E * 32;

    for (int l = 0; l < 6; ++l) {
        float* layer_out = ping[l & 1];
        gcn_zero_kernel<<<4096, ZB, 0, stream>>>(AGG, GCN_N * GCN_D);
        gcn_scatter_kernel<<<(scatter_threads + ZB - 1) / ZB, ZB, 0, stream>>>(
            layer_in, normS, src, dst, AGG, GCN_E);
        gcn_gemm_relu_kernel<<<gemm_grid, 32, 0, stream>>>(
            AGG, normD, Wl[l], bl[l], layer_out);
        layer_in = layer_out;
    }

    // --- mean pooling + MLP head -------------------------------------------
    gcn_zero_kernel<<<512, ZB, 0, stream>>>(pooled, GCN_G * GCN_D);
    gcn_zero_kernel<<<8, ZB, 0, stream>>>(cnt, GCN_G);
    gcn_pool_kernel<<<(GCN_N * 32 + ZB - 1) / ZB, ZB, 0, stream>>>(
        layer_in, graph_id, pooled, cnt, GCN_N);
    gcn_head1_kernel<<<(GCN_G * 64 + ZB - 1) / ZB, ZB, 0, stream>>>(
        pooled, cnt, Wp1, bp1, hidden);
    gcn_head2_kernel<<<(GCN_G * GCN_NC + ZB - 1) / ZB, ZB, 0, stream>>>(
        hidden, Wp2, bp2, scores);
}